// DGLGraphConv_64132451664452
// MI455X (gfx1250) — compile-verified
//
#include <hip/hip_runtime.h>
#include <math.h>

typedef __attribute__((ext_vector_type(2))) float v2f;
typedef __attribute__((ext_vector_type(8))) float v8f;
typedef __attribute__((ext_vector_type(4))) int   v4i;
typedef __attribute__((address_space(1))) v4i g1v4i;   // global int4
typedef __attribute__((address_space(3))) v4i l3v4i;   // LDS int4

#define IN_F   256
#define OUT_F  128
#define RANK   64
#define A_LDS  260   // padded row stride (floats): row m -> bank (4m)%64, conflict-free

// ---------------------------------------------------------------------------
// Zero-fill helper (graph-capture safe)
// ---------------------------------------------------------------------------
__global__ __launch_bounds__(256) void zero_f32(float* __restrict__ p, size_t n) {
    size_t i = (size_t)blockIdx.x * blockDim.x + threadIdx.x;
    size_t stride = (size_t)gridDim.x * blockDim.x;
    for (; i < n; i += stride) p[i] = 0.0f;
}

// ---------------------------------------------------------------------------
// Phase 1: fused GEMM  C[N,192] = feat[N,256] @ [W_sum | W_prod]
//   cols 0..127   -> h_sum  (raw)
//   cols 128..191 -> h_prod (tanh applied)
// Block = 128 threads (4 waves). Block owns a 32-row M tile (two 16-row WMMA
// tiles), staged in LDS via async-to-LDS when available. Each wave covers 3 of
// the 12 N-tiles; B registers are reused across both M-tiles (2 wmma per
// B-load pair) with V_WMMA_F32_16X16X4_F32.
// A layout: a[r] = A[lane%16][2*(lane/16)+r]      (r = 0,1)
// B layout: b[r] = B[2*(lane/16)+r][lane%16]
// C layout: c[r] = C[r + 8*(lane/16)][lane%16]    (r = 0..7)
// ---------------------------------------------------------------------------
__global__ __launch_bounds__(128) void phase1_gemm(
    const float* __restrict__ feat,
    const float* __restrict__ wsum,   // [256,128]
    const float* __restrict__ wprod,  // [256,64]
    float* __restrict__ hsum,         // [N,128]
    float* __restrict__ hprod)        // [N,64]
{
    __shared__ float Atile[32 * A_LDS];

    const int tid  = threadIdx.x;
    const int lane = tid & 31;
    const int wave = tid >> 5;           // 0..3
    const int m0   = blockIdx.x * 32;

    // Stage 32x256 A tile: 2048 16-byte chunks, 16 per thread.
    // A_LDS*4 = 1040 bytes per row (16B aligned), cols in 16B steps.
    for (int idx = tid; idx < 32 * (IN_F / 4); idx += 128) {
        const int row = idx >> 6;              // 0..31
        const int c4  = (idx & 63) << 2;       // 0,4,...,252
        const float* g = feat + (size_t)(m0 + row) * IN_F + c4;
        float*       l = &Atile[row * A_LDS + c4];
#if __has_builtin(__builtin_amdgcn_global_load_async_to_lds_b128)
        __builtin_amdgcn_global_load_async_to_lds_b128(
            (g1v4i*)g, (l3v4i*)l, /*offset=*/0, /*cpol=*/0);
#else
        *(float4*)l = *(const float4*)g;
#endif
    }
#if __has_builtin(__builtin_amdgcn_global_load_async_to_lds_b128)
#if __has_builtin(__builtin_amdgcn_s_wait_asynccnt)
    __builtin_amdgcn_s_wait_asynccnt(0);
#else
    asm volatile("s_wait_asynccnt 0" ::: "memory");
#endif
#endif
    __syncthreads();

    const int mrow  = lane & 15;          // row-in-tile for A, col-in-tile for B/C
    const int khalf = (lane >> 4) << 1;   // 0 or 2

    for (int t = 0; t < 3; ++t) {
        const int nt = wave * 3 + t;      // 0..11
        const float* W;
        int ldw, col;
        if (nt < 8) { W = wsum;  ldw = OUT_F; col = nt * 16 + mrow; }
        else        { W = wprod; ldw = RANK;  col = (nt - 8) * 16 + mrow; }

        v8f c0 = {}, c1 = {};
        for (int kb = 0; kb < IN_F; kb += 4) {
            v2f b, a0, a1;
            b.x  = W[(size_t)(kb + khalf)     * ldw + col];
            b.y  = W[(size_t)(kb + khalf + 1) * ldw + col];
            a0.x = Atile[mrow * A_LDS + kb + khalf];
            a0.y = Atile[mrow * A_LDS + kb + khalf + 1];
            a1.x = Atile[(16 + mrow) * A_LDS + kb + khalf];
            a1.y = Atile[(16 + mrow) * A_LDS + kb + khalf + 1];
            c0 = __builtin_amdgcn_wmma_f32_16x16x4_f32(
                     false, a0, false, b, (short)0, c0, false, false);
            c1 = __builtin_amdgcn_wmma_f32_16x16x4_f32(
                     false, a1, false, b, (short)0, c1, false, false);
        }

        const int mb0 = m0      + ((lane >> 4) << 3);
        const int mb1 = m0 + 16 + ((lane >> 4) << 3);
        if (nt < 8) {
            const int ccol = nt * 16 + mrow;
            #pragma unroll
            for (int r = 0; r < 8; ++r) {
                hsum[(size_t)(mb0 + r) * OUT_F + ccol] = c0[r];
                hsum[(size_t)(mb1 + r) * OUT_F + ccol] = c1[r];
            }
        } else {
            const int ccol = (nt - 8) * 16 + mrow;
            #pragma unroll
            for (int r = 0; r < 8; ++r) {
                hprod[(size_t)(mb0 + r) * RANK + ccol] = tanhf(c0[r]);
                hprod[(size_t)(mb1 + r) * RANK + ccol] = tanhf(c1[r]);
            }
        }
    }
}

// ---------------------------------------------------------------------------
// Phase 2: edge scatter. One wave per edge (grid-stride).
//   sum_agg  += h_sum[src]            (atomic f32 add, into d_out)
//   logsum   += log|h_prod[src]|      (atomic f32 add; log(0) = -inf => prod 0)
//   signpar  ^= (h_prod[src] < 0)     (atomic xor parity)
// ---------------------------------------------------------------------------
__global__ __launch_bounds__(256) void phase2_edges(
    const int*   __restrict__ src,
    const int*   __restrict__ dst,
    const float* __restrict__ hsum,
    const float* __restrict__ hprod,
    float*       __restrict__ sumagg,   // d_out, pre-zeroed
    float*       __restrict__ logsum,   // pre-zeroed
    unsigned*    __restrict__ signpar,  // pre-zeroed
    int n_edges)
{
    const int lane    = threadIdx.x & 31;
    const int waveIdx = (blockIdx.x * (blockDim.x >> 5)) + (threadIdx.x >> 5);
    const int nWaves  = (gridDim.x * blockDim.x) >> 5;

    for (int e = waveIdx; e < n_edges; e += nWaves) {
        const int s = src[e];
        const int d = dst[e];

        const float* hs = hsum + (size_t)s * OUT_F;
        float*       sa = sumagg + (size_t)d * OUT_F;
        #pragma unroll
        for (int j = 0; j < OUT_F / 32; ++j)
            atomicAdd(&sa[lane + 32 * j], hs[lane + 32 * j]);

        const float* hp = hprod + (size_t)s * RANK;
        float*       ls = logsum + (size_t)d * RANK;
        unsigned*    sp = signpar + (size_t)d * RANK;
        #pragma unroll
        for (int j = 0; j < RANK / 32; ++j) {
            float p = hp[lane + 32 * j];
            atomicAdd(&ls[lane + 32 * j], logf(fabsf(p)));
            if (p < 0.0f) atomicXor(&sp[lane + 32 * j], 1u);
        }
    }
}

// ---------------------------------------------------------------------------
// Phase 3: one wave per node.
//   prod_nb  = sign * exp(logsum)     (empty segment -> exp(0) = 1, correct)
//   prod_agg = prod_nb @ v            (v staged in LDS, 32KB)
//   attention blend, overwrite d_out (which currently holds sum_agg)
// ---------------------------------------------------------------------------
__global__ __launch_bounds__(256) void phase3_final(
    const float*    __restrict__ logsum,
    const unsigned* __restrict__ signpar,
    const float*    __restrict__ v,       // [64,128]
    const float*    __restrict__ att1,    // [128]
    const float*    __restrict__ att2,    // [128]
    const float*    __restrict__ attvec,  // [2,2] row-major
    float*          __restrict__ out,     // holds sum_agg, overwritten
    int n_nodes)
{
    __shared__ float Vs[RANK * OUT_F];    // 32 KB
    __shared__ float pn[8][RANK];         // 2 KB

    const int tid  = threadIdx.x;
    const int lane = tid & 31;
    const int wave = tid >> 5;

    for (int i = tid; i < RANK * OUT_F; i += 256) Vs[i] = v[i];
    __syncthreads();

    const int n = blockIdx.x * 8 + wave;
    const bool active = (n < n_nodes);

    if (active) {
        #pragma unroll
        for (int j = 0; j < RANK / 32; ++j) {
            int k = lane + 32 * j;
            float e = expf(logsum[(size_t)n * RANK + k]);
            pn[wave][k] = (signpar[(size_t)n * RANK + k] & 1u) ? -e : e;
        }
    }
    __syncthreads();

    if (!active) return;

    float pa[4], sa[4];
    float l1 = 0.0f, l2 = 0.0f;
    #pragma unroll
    for (int j = 0; j < 4; ++j) {
        const int c = lane + 32 * j;
        float acc = 0.0f;
        #pragma unroll 8
        for (int k = 0; k < RANK; ++k)
            acc = fmaf(pn[wave][k], Vs[k * OUT_F + c], acc);
        pa[j] = acc;
        sa[j] = out[(size_t)n * OUT_F + c];
        l1 = fmaf(acc,   att1[c], l1);
        l2 = fmaf(sa[j], att2[c], l2);
    }
    // wave32 reduction
    #pragma unroll
    for (int off = 16; off >= 1; off >>= 1) {
        l1 += __shfl_xor(l1, off, 32);
        l2 += __shfl_xor(l2, off, 32);
    }

    const float s1 = 1.0f / (1.0f + expf(-l1));
    const float s2 = 1.0f / (1.0f + expf(-l2));
    // z = sigmoid(logits) @ att_vec_w.T ; att = softmax(z / 2)
    const float q0 = 0.5f * (s1 * attvec[0] + s2 * attvec[1]);
    const float q1 = 0.5f * (s1 * attvec[2] + s2 * attvec[3]);
    const float m  = fmaxf(q0, q1);
    const float e0 = expf(q0 - m);
    const float e1 = expf(q1 - m);
    const float inv = 1.0f / (e0 + e1);
    const float a0 = e0 * inv;
    const float a1 = e1 * inv;

    #pragma unroll
    for (int j = 0; j < 4; ++j) {
        const int c = lane + 32 * j;
        out[(size_t)n * OUT_F + c] = a0 * pa[j] + a1 * sa[j];
    }
}

// ---------------------------------------------------------------------------
extern "C" void kernel_launch(void* const* d_in, const int* in_sizes, int n_in,
                              void* d_out, int out_size, void* d_ws, size_t ws_size,
                              hipStream_t stream) {
    const float* feat   = (const float*)d_in[0];
    const int*   src    = (const int*)  d_in[1];
    const int*   dst    = (const int*)  d_in[2];
    const float* wsum   = (const float*)d_in[3];
    const float* wprod  = (const float*)d_in[4];
    const float* v      = (const float*)d_in[5];
    const float* att1   = (const float*)d_in[6];
    const float* att2   = (const float*)d_in[7];
    const float* attvec = (const float*)d_in[8];
    float* out = (float*)d_out;

    const int nNodes = in_sizes[0] / IN_F;   // 100000
    const int nEdges = in_sizes[1];          // 1600000

    // Workspace layout (total 128 MB for N=100k):
    //   hsum   : N*128 f32
    //   hprod  : N*64  f32
    //   logsum : N*64  f32
    //   signpar: N*64  u32
    char* ws = (char*)d_ws;
    float*    hsum    = (float*)ws;
    float*    hprod   = (float*)(ws + (size_t)nNodes * OUT_F * 4);
    float*    logsum  = (float*)(ws + (size_t)nNodes * (OUT_F + RANK) * 4);
    unsigned* signpar = (unsigned*)(ws + (size_t)nNodes * (OUT_F + 2 * RANK) * 4);

    // Zero accumulators (sum_agg lives in d_out; signpar zeroed as float-0 bits)
    zero_f32<<<2048, 256, 0, stream>>>(out, (size_t)nNodes * OUT_F);
    zero_f32<<<2048, 256, 0, stream>>>(logsum, (size_t)nNodes * RANK);
    zero_f32<<<2048, 256, 0, stream>>>((float*)signpar, (size_t)nNodes * RANK);

    // Phase 1: WMMA GEMM (100000/32 = 3125 blocks, 32 rows each)
    phase1_gemm<<<nNodes / 32, 128, 0, stream>>>(feat, wsum, wprod, hsum, hprod);

    // Phase 2: edge scatter (atomic adds / xors)
    phase2_edges<<<4096, 256, 0, stream>>>(src, dst, hsum, hprod,
                                           out, logsum, signpar, nEdges);

    // Phase 3: per-node blend (100000/8 waves-per-block = 12500 blocks)
    phase3_final<<<(nNodes + 7) / 8, 256, 0, stream>>>(logsum, signpar, v,
                                                       att1, att2, attvec,
                                                       out, nNodes);
}